// MetaLearner_53687091200293
// MI455X (gfx1250) — compile-verified
//
#include <hip/hip_runtime.h>
#include <math.h>

// ---------------- problem constants (match reference) ----------------
#define BQ    4
#define NQ    2048
#define CH    128
#define NSUP  2048
#define TOPKK 512
#define QROWS 2048           // BQ*TOPKK
#define NN    4096           // QROWS + NSUP
#define KNN1  11             // KNN+1 (incl. self)
#define PD    300
#define PDP   304            // PD padded to multiple of 16
#define OD    8
#define LPA   0.99f
#define CGIT  64

typedef __attribute__((ext_vector_type(2))) float v2f;
typedef __attribute__((ext_vector_type(8))) float v8f;

static __device__ __forceinline__ v8f wmma_f32(v2f a, v2f b, v8f c) {
    return __builtin_amdgcn_wmma_f32_16x16x4_f32(false, a, false, b, (short)0, c,
                                                 false, false);
}

// ---------------- 1. importance MLP: 128->64->64->1, sigmoid ----------------
__global__ void k_importance(const float* __restrict__ h, const float* __restrict__ w1,
                             const float* __restrict__ b1, const float* __restrict__ w2,
                             const float* __restrict__ b2, const float* __restrict__ wa,
                             const float* __restrict__ ba, float* __restrict__ imp) {
    int row = blockIdx.x;          // 0..BQ*NQ-1
    int t   = threadIdx.x;         // 0..63
    __shared__ float sh[CH];
    __shared__ float sf[64];
    sh[t]      = h[(size_t)row * CH + t];
    sh[t + 64] = h[(size_t)row * CH + t + 64];
    __syncthreads();
    float a = b1[t];
    #pragma unroll 8
    for (int k = 0; k < CH; ++k) a += sh[k] * w1[k * 64 + t];
    a = a > 0.f ? a : 0.1f * a;
    sf[t] = a;
    __syncthreads();
    float a2 = b2[t];
    #pragma unroll 8
    for (int k = 0; k < 64; ++k) a2 += sf[k] * w2[k * 64 + t];
    a2 = a2 > 0.f ? a2 : 0.1f * a2;
    __syncthreads();
    sf[t] = a2 * wa[t];
    __syncthreads();
    for (int s = 32; s > 0; s >>= 1) { if (t < s) sf[t] += sf[t + s]; __syncthreads(); }
    if (t == 0) {
        float z = sf[0] + ba[0];
        imp[row] = 1.f / (1.f + expf(-z));
    }
}

// ---------------- 2. top-512 per batch via stable rank selection ----------------
__global__ void k_topk(const float* __restrict__ imp, int* __restrict__ inds) {
    int b = blockIdx.x, t = threadIdx.x;        // 256 threads
    __shared__ float v[NQ];
    for (int i = t; i < NQ; i += 256) v[i] = imp[b * NQ + i];
    __syncthreads();
    for (int i = t; i < NQ; i += 256) {
        float vi = v[i];
        int rank = 0;
        for (int j = 0; j < NQ; ++j) {
            float vj = v[j];
            rank += (vj > vi) || (vj == vi && j < i);
        }
        if (rank < TOPKK) inds[b * TOPKK + rank] = i;   // descending, stable ties
    }
}

// ---------------- 3. gather f = [h_sub ; support], row squared norms ----------------
__global__ void k_buildf(const float* __restrict__ h, const float* __restrict__ se,
                         const int* __restrict__ inds, float* __restrict__ f,
                         float* __restrict__ sq) {
    int row = blockIdx.x;          // 0..NN-1
    int t   = threadIdx.x;         // 0..127
    float val;
    if (row < QROWS) {
        int b = row / TOPKK, k = row % TOPKK;
        int src = inds[b * TOPKK + k];
        val = h[((size_t)(b * NQ + src)) * CH + t];
    } else {
        val = se[(size_t)(row - QROWS) * CH + t];
    }
    f[(size_t)row * CH + t] = val;
    __shared__ float s[128];
    s[t] = val * val;
    __syncthreads();
    for (int st = 64; st > 0; st >>= 1) { if (t < st) s[t] += s[t + st]; __syncthreads(); }
    if (t == 0) sq[row] = s[0];
}

// ---------------- 4. pairwise squared L2 via WMMA f32 Gram ----------------
// block = 8 waves; wave w -> 16x16 tile; block covers 128 rows x 16 cols
__global__ void k_d2(const float* __restrict__ f, const float* __restrict__ sq,
                     float* __restrict__ d2) {
    int wave = threadIdx.x >> 5, lane = threadIdx.x & 31;
    int l16 = lane & 15, hh = lane >> 4;
    int m0 = blockIdx.x * 128 + wave * 16;
    int n0 = blockIdx.y * 16;
    const float* arow = f + (size_t)(m0 + l16) * CH + 2 * hh;
    const float* brow = f + (size_t)(n0 + l16) * CH + 2 * hh;   // B = f^T, K-pair contiguous
    v8f acc = {};
    #pragma unroll 4
    for (int k0 = 0; k0 < CH; k0 += 4) {
        v2f a = *(const v2f*)(arow + k0);
        v2f b = *(const v2f*)(brow + k0);
        acc = wmma_f32(a, b, acc);
    }
    #pragma unroll
    for (int r = 0; r < 8; ++r) {
        int row = m0 + r + 8 * hh, col = n0 + l16;
        float v = sq[row] + sq[col] - 2.0f * acc[r];
        d2[(size_t)row * NN + col] = v > 0.f ? v : 0.f;
    }
}

// ---------------- 5. exact 11-NN per row (ascending d2, stable ties) ----------------
__global__ void k_knn(const float* __restrict__ d2, int* __restrict__ Ik,
                      float* __restrict__ Aff) {
    int i = blockIdx.x * 64 + threadIdx.x;
    if (i >= NN) return;
    float bd[KNN1]; int bi[KNN1];
    #pragma unroll
    for (int c = 0; c < KNN1; ++c) { bd[c] = 3.0e38f; bi[c] = NN; }
    const float* row = d2 + (size_t)i * NN;
    for (int j = 0; j < NN; ++j) {
        float d = row[j];
        if (d < bd[KNN1 - 1] || (d == bd[KNN1 - 1] && j < bi[KNN1 - 1])) {
            int c = KNN1 - 1; bd[c] = d; bi[c] = j;
            while (c > 0 && (bd[c] < bd[c - 1] || (bd[c] == bd[c - 1] && bi[c] < bi[c - 1]))) {
                float td = bd[c]; bd[c] = bd[c - 1]; bd[c - 1] = td;
                int  ti = bi[c]; bi[c] = bi[c - 1]; bi[c - 1] = ti;
                --c;
            }
        }
    }
    #pragma unroll
    for (int c = 0; c < KNN1; ++c) Ik[i * KNN1 + c] = bi[c];
    #pragma unroll
    for (int c = 1; c < KNN1; ++c) Aff[i * (KNN1 - 1) + (c - 1)] = expf(-bd[c]); // sigma=1
}

// ---------------- 6. zero the n x n buffer ----------------
__global__ void k_zero(float* __restrict__ p, unsigned long n) {
    unsigned long i = (unsigned long)blockIdx.x * blockDim.x + threadIdx.x;
    if (i < n) p[i] = 0.f;
}

// ---------------- 7. scatter symmetric affinities: W = scatter + scatter^T --------
// each cell receives <=2 commutative float adds -> bit-deterministic
__global__ void k_scatter(const int* __restrict__ Ik, const float* __restrict__ Aff,
                          float* __restrict__ W) {
    int i = blockIdx.x * 64 + threadIdx.x;
    if (i >= NN) return;
    #pragma unroll
    for (int c = 1; c < KNN1; ++c) {
        int col = Ik[i * KNN1 + c];
        if (col >= NN) col = NN - 1;                    // clip like reference
        float val = Aff[i * (KNN1 - 1) + (c - 1)];
        atomicAdd(&W[(size_t)i * NN + col], val);
        atomicAdd(&W[(size_t)col * NN + i], val);
    }
}

// ---------------- 8. degree d = 1/sqrt(rowsum excluding diagonal) ----------------
__global__ void k_rowsum(const float* __restrict__ W, float* __restrict__ dd) {
    int i = blockIdx.x, t = threadIdx.x;               // 256 threads
    const float* row = W + (size_t)i * NN;
    float s = 0.f;
    for (int j = t; j < NN; j += 256) s += row[j];
    __shared__ float sh[256];
    sh[t] = s; __syncthreads();
    for (int st = 128; st > 0; st >>= 1) { if (t < st) sh[t] += sh[t + st]; __syncthreads(); }
    if (t == 0) {
        float S = sh[0] - row[i];                      // zeroed diagonal
        if (S == 0.f) S = 1.f;
        dd[i] = rsqrtf(S);
    }
}

// ---------------- 9. A = I - alpha * D^-1/2 W D^-1/2 (in place over W) ----------
__global__ void k_buildA(float* __restrict__ W, const float* __restrict__ dd) {
    size_t idx = (size_t)blockIdx.x * blockDim.x + threadIdx.x;
    if (idx >= (size_t)NN * NN) return;
    int i = (int)(idx >> 12), j = (int)(idx & (NN - 1));
    float w = (i == j) ? 0.f : W[idx];
    W[idx] = (i == j ? 1.f : 0.f) - LPA * dd[i] * dd[j] * w;
}

// ---------------- 10. CG init: X=0, R=P=Y (Y = [0 ; s], padded cols zero) --------
__global__ void k_cginit(const float* __restrict__ s, float* __restrict__ X,
                         float* __restrict__ R, float* __restrict__ P) {
    size_t idx = (size_t)blockIdx.x * blockDim.x + threadIdx.x;
    if (idx >= (size_t)NN * PDP) return;
    int i = (int)(idx / PDP), j = (int)(idx % PDP);
    float y = (i >= QROWS && j < PD) ? s[(size_t)(i - QROWS) * PD + j] : 0.f;
    X[idx] = 0.f; R[idx] = y; P[idx] = y;
}

// ---------------- 11. AP = A @ P via WMMA f32 (K = 4096) -------------------------
// block = 8 waves, block tile 512 rows x 16 cols; wave owns 4 row-tiles (64x16)
// B chunk (64 k x 16 n) staged in LDS transposed so a lane's K-pair is one b64.
__global__ void k_gemmAP(const float* __restrict__ A, const float* __restrict__ P,
                         float* __restrict__ AP) {
    __shared__ float Bs[16 * 64];                      // [col][k], 4 KB
    int wave = threadIdx.x >> 5, lane = threadIdx.x & 31;
    int l16 = lane & 15, hh = lane >> 4;
    int m0 = blockIdx.x * 512 + wave * 64;
    int n0 = blockIdx.y * 16;
    const float* a0 = A + (size_t)(m0 + l16) * NN + 2 * hh;
    v8f acc0 = {}, acc1 = {}, acc2 = {}, acc3 = {};
    for (int k0 = 0; k0 < NN; k0 += 64) {
        __syncthreads();
        for (int idx = threadIdx.x; idx < 1024; idx += 256) {
            int col = idx >> 6, kk = idx & 63;
            Bs[col * 64 + kk] = P[(size_t)(k0 + kk) * PDP + n0 + col];
        }
        __syncthreads();
        // speculative prefetch of next A chunk (dropped harmlessly at the tail)
        __builtin_prefetch(a0 + k0 + 64, 0, 0);
        __builtin_prefetch(a0 + (size_t)16 * NN + k0 + 64, 0, 0);
        __builtin_prefetch(a0 + (size_t)32 * NN + k0 + 64, 0, 0);
        __builtin_prefetch(a0 + (size_t)48 * NN + k0 + 64, 0, 0);
        const float* bs = Bs + l16 * 64 + 2 * hh;
        #pragma unroll
        for (int kk = 0; kk < 64; kk += 4) {
            v2f b = *(const v2f*)(bs + kk);
            v2f a;
            a = *(const v2f*)(a0 + k0 + kk);                        acc0 = wmma_f32(a, b, acc0);
            a = *(const v2f*)(a0 + (size_t)16 * NN + k0 + kk);      acc1 = wmma_f32(a, b, acc1);
            a = *(const v2f*)(a0 + (size_t)32 * NN + k0 + kk);      acc2 = wmma_f32(a, b, acc2);
            a = *(const v2f*)(a0 + (size_t)48 * NN + k0 + kk);      acc3 = wmma_f32(a, b, acc3);
        }
    }
    int col = n0 + l16;
    #pragma unroll
    for (int r = 0; r < 8; ++r) {
        int mr = m0 + r + 8 * hh;
        AP[(size_t)(mr +  0) * PDP + col] = acc0[r];
        AP[(size_t)(mr + 16) * PDP + col] = acc1[r];
        AP[(size_t)(mr + 32) * PDP + col] = acc2[r];
        AP[(size_t)(mr + 48) * PDP + col] = acc3[r];
    }
}

// ---------------- 12. per-column dot products (one block per column) -------------
__global__ void k_coldot(const float* __restrict__ Am, const float* __restrict__ Bm,
                         float* __restrict__ out) {
    int j = blockIdx.x, t = threadIdx.x;               // 256 threads
    float s = 0.f;
    for (int i = t; i < NN; i += 256)
        s += Am[(size_t)i * PDP + j] * Bm[(size_t)i * PDP + j];
    __shared__ float sh[256];
    sh[t] = s; __syncthreads();
    for (int st = 128; st > 0; st >>= 1) { if (t < st) sh[t] += sh[t + st]; __syncthreads(); }
    if (t == 0) out[j] = sh[0];
}

__global__ void k_alpha(const float* __restrict__ rs, const float* __restrict__ pap,
                        float* __restrict__ alf) {
    int j = blockIdx.x * 64 + threadIdx.x;
    if (j < PDP) {
        float d = pap[j], r = rs[j];
        alf[j] = (d != 0.f && r != 0.f) ? r / d : 0.f;
    }
}

__global__ void k_updxr(float* __restrict__ X, float* __restrict__ R,
                        const float* __restrict__ P, const float* __restrict__ AP,
                        const float* __restrict__ alf) {
    size_t idx = (size_t)blockIdx.x * blockDim.x + threadIdx.x;
    if (idx >= (size_t)NN * PDP) return;
    float a = alf[idx % PDP];
    X[idx] += a * P[idx];
    R[idx] -= a * AP[idx];
}

__global__ void k_beta(float* __restrict__ rs, const float* __restrict__ rsn,
                       float* __restrict__ bet) {
    int j = blockIdx.x * 64 + threadIdx.x;
    if (j < PDP) {
        float o = rs[j], n = rsn[j];
        bet[j] = (o != 0.f) ? n / o : 0.f;
        rs[j] = n;
    }
}

__global__ void k_updp(float* __restrict__ P, const float* __restrict__ R,
                       const float* __restrict__ bet) {
    size_t idx = (size_t)blockIdx.x * blockDim.x + threadIdx.x;
    if (idx >= (size_t)NN * PDP) return;
    P[idx] = R[idx] + bet[idx % PDP] * P[idx];
}

// ---------------- 13. final FC: out[row,o] = Z[row,:300] @ fc_w + fc_b ----------
__global__ void k_fc(const float* __restrict__ X, const float* __restrict__ fcw,
                     const float* __restrict__ fcb, float* __restrict__ out) {
    int idx = blockIdx.x * 256 + threadIdx.x;          // QROWS*OD = 16384
    if (idx >= QROWS * OD) return;
    int row = idx / OD, o = idx % OD;
    float acc = fcb[o];
    #pragma unroll 4
    for (int p = 0; p < PD; ++p) acc += X[(size_t)row * PDP + p] * fcw[p * OD + o];
    out[idx] = acc;
}

// =================================================================================
extern "C" void kernel_launch(void* const* d_in, const int* in_sizes, int n_in,
                              void* d_out, int out_size, void* d_ws, size_t ws_size,
                              hipStream_t stream) {
    (void)in_sizes; (void)n_in; (void)out_size; (void)ws_size;
    const float* h   = (const float*)d_in[0];
    const float* se  = (const float*)d_in[1];
    const float* s   = (const float*)d_in[2];
    const float* w1  = (const float*)d_in[3];
    const float* b1  = (const float*)d_in[4];
    const float* w2  = (const float*)d_in[5];
    const float* b2  = (const float*)d_in[6];
    const float* wa  = (const float*)d_in[7];
    const float* ba  = (const float*)d_in[8];
    const float* fcw = (const float*)d_in[9];
    const float* fcb = (const float*)d_in[10];
    float* out = (float*)d_out;

    // workspace bump allocator (~90 MB total)
    char* p = (char*)d_ws;
    auto alloc = [&](size_t bytes) -> void* {
        void* r = (void*)p; p += (bytes + 255) & ~(size_t)255; return r;
    };
    float* Abuf = (float*)alloc(sizeof(float) * (size_t)NN * NN);   // d2 -> W -> A
    float* X    = (float*)alloc(sizeof(float) * (size_t)NN * PDP);
    float* R    = (float*)alloc(sizeof(float) * (size_t)NN * PDP);
    float* Pm   = (float*)alloc(sizeof(float) * (size_t)NN * PDP);
    float* AP   = (float*)alloc(sizeof(float) * (size_t)NN * PDP);
    float* f    = (float*)alloc(sizeof(float) * (size_t)NN * CH);
    float* sq   = (float*)alloc(sizeof(float) * NN);
    float* imp  = (float*)alloc(sizeof(float) * BQ * NQ);
    int*   inds = (int*)  alloc(sizeof(int) * BQ * TOPKK);
    int*   Ik   = (int*)  alloc(sizeof(int) * NN * KNN1);
    float* Aff  = (float*)alloc(sizeof(float) * NN * (KNN1 - 1));
    float* dd   = (float*)alloc(sizeof(float) * NN);
    float* rs   = (float*)alloc(sizeof(float) * PDP);
    float* rsn  = (float*)alloc(sizeof(float) * PDP);
    float* pap  = (float*)alloc(sizeof(float) * PDP);
    float* alf  = (float*)alloc(sizeof(float) * PDP);
    float* bet  = (float*)alloc(sizeof(float) * PDP);

    const size_t nnTot = (size_t)NN * NN;       // 16M
    const size_t npTot = (size_t)NN * PDP;      // 1.25M
    const int nnBlocks = (int)((nnTot + 255) / 256);
    const int npBlocks = (int)((npTot + 255) / 256);

    // stage 1: importance + top-k + gather
    k_importance<<<BQ * NQ, 64, 0, stream>>>(h, w1, b1, w2, b2, wa, ba, imp);
    k_topk<<<BQ, 256, 0, stream>>>(imp, inds);
    k_buildf<<<NN, 128, 0, stream>>>(h, se, inds, f, sq);

    // stage 2: pairwise d2 (WMMA Gram) + exact kNN
    k_d2<<<dim3(NN / 128, NN / 16), 256, 0, stream>>>(f, sq, Abuf);
    k_knn<<<NN / 64, 64, 0, stream>>>(Abuf, Ik, Aff);

    // stage 3: graph operator A = I - alpha * D^-1/2 W D^-1/2
    k_zero<<<nnBlocks, 256, 0, stream>>>(Abuf, (unsigned long)nnTot);
    k_scatter<<<NN / 64, 64, 0, stream>>>(Ik, Aff, Abuf);
    k_rowsum<<<NN, 256, 0, stream>>>(Abuf, dd);
    k_buildA<<<nnBlocks, 256, 0, stream>>>(Abuf, dd);

    // stage 4: A symmetric SPD  =>  inv(A^T A) A^T Y == A^{-1} Y : block CG
    k_cginit<<<npBlocks, 256, 0, stream>>>(s, X, R, Pm);
    k_coldot<<<PDP, 256, 0, stream>>>(R, R, rs);
    for (int it = 0; it < CGIT; ++it) {
        k_gemmAP<<<dim3(NN / 512, PDP / 16), 256, 0, stream>>>(Abuf, Pm, AP);
        k_coldot<<<PDP, 256, 0, stream>>>(Pm, AP, pap);
        k_alpha<<<(PDP + 63) / 64, 64, 0, stream>>>(rs, pap, alf);
        k_updxr<<<npBlocks, 256, 0, stream>>>(X, R, Pm, AP, alf);
        k_coldot<<<PDP, 256, 0, stream>>>(R, R, rsn);
        k_beta<<<(PDP + 63) / 64, 64, 0, stream>>>(rs, rsn, bet);
        k_updp<<<npBlocks, 256, 0, stream>>>(Pm, R, bet);
    }

    // stage 5: final FC on Z[:q]
    k_fc<<<(QROWS * OD + 255) / 256, 256, 0, stream>>>(X, fcw, fcb, out);
}